// SuperGAT_62904091018059
// MI455X (gfx1250) — compile-verified
//
#include <hip/hip_runtime.h>
#include <hip/hip_bf16.h>
#include <math.h>

// ---------------------------------------------------------------------------
// SuperGAT forward for MI455X (gfx1250), wave32, f32 WMMA 16x16x4,
// LDS-staged A tiles (bank-conflict-padded), global prefetch.
// ---------------------------------------------------------------------------

#define N_NODES   20000
#define N_EDGES   160000
#define N_IN      300
#define NHID      128
#define HEADS     4
#define NOUT      768
#define NGRAPHS   128

#define BK        64              // K-chunk staged through LDS
#define LDS_STR   (BK + 2)        // padded row stride (floats): banks 2m..2m+1

typedef __attribute__((ext_vector_type(2))) float v2f;
typedef __attribute__((ext_vector_type(8))) float v8f;

// Order-preserving float <-> uint encoding for atomic max on floats.
__device__ __forceinline__ unsigned fenc(float f) {
    unsigned u = __float_as_uint(f);
    return (u & 0x80000000u) ? ~u : (u | 0x80000000u);
}
__device__ __forceinline__ float fdec(unsigned u) {
    return (u & 0x80000000u) ? __uint_as_float(u & 0x7fffffffu)
                             : __uint_as_float(~u);
}
#define ENC_NEG_INF 0x007FFFFFu   // fenc(-inf)

// ---------------------------------------------------------------------------
// fills
// ---------------------------------------------------------------------------
__global__ void fill_f32_kernel(float* p, float v, int n) {
    int i = blockIdx.x * blockDim.x + threadIdx.x;
    if (i < n) p[i] = v;
}
__global__ void fill_u32_kernel(unsigned* p, unsigned v, int n) {
    int i = blockIdx.x * blockDim.x + threadIdx.x;
    if (i < n) p[i] = v;
}

// ---------------------------------------------------------------------------
// WMMA f32 GEMM: C[M,N] = act(A[M,K] @ B[K,N] + bias)
// grid = (M/16, (N/16)/8), block = 256 (8 waves; one 16x16 tile per wave).
// A tile (16 x BK) staged through LDS once per block (coalesced float4),
// shared by all 8 waves.  B loads stay global (coalesced per half-wave).
// ---------------------------------------------------------------------------
__global__ void gemm16_wmma(const float* __restrict__ A,
                            const float* __restrict__ B,
                            const float* __restrict__ bias,
                            float* __restrict__ C,
                            int M, int N, int K, float slope, int act) {
    __shared__ float As[16 * LDS_STR];

    const int tid  = threadIdx.x;
    const int lane = tid & 31;
    const int wave = tid >> 5;
    const int row0 = blockIdx.x * 16;
    const int col0 = (blockIdx.y * 8 + wave) * 16;
    const int m    = lane & 15;        // A row / B col / D col within tile
    const int half = lane >> 4;        // K sub-slot select
    const int kb   = half * 2;

    // cooperative A-tile loader: 256 threads x float4 = 16 x 64 floats
    const int lrow = tid >> 4;         // 0..15
    const int lcol = (tid & 15) * 4;   // 0,4,...,60
    const float* Aload = A + (size_t)(row0 + lrow) * K + lcol;

    const float* Bcol = B + col0 + m;

    v8f acc = {0.f, 0.f, 0.f, 0.f, 0.f, 0.f, 0.f, 0.f};

    for (int k0 = 0; k0 < K; k0 += BK) {
        const int bk = (K - k0) < BK ? (K - k0) : BK;
        __syncthreads();
        if (lcol < bk) {
            const float4 av = *(const float4*)(Aload + k0);
            float* s = &As[lrow * LDS_STR + lcol];
            s[0] = av.x; s[1] = av.y; s[2] = av.z; s[3] = av.w;
            if (k0 + BK < K)   // prefetch next A chunk into cache
                __builtin_prefetch(Aload + k0 + BK, 0, 0);
        }
        __syncthreads();

        const float* Arow = &As[m * LDS_STR + kb];
        #pragma unroll 4
        for (int kk = 0; kk < bk; kk += 4) {
            v2f a = *(const v2f*)(Arow + kk);              // A[m][k+kb], A[m][k+kb+1]
            v2f b;
            b.x = Bcol[(size_t)(k0 + kk + kb) * N];        // B[k+kb  ][n]
            b.y = Bcol[(size_t)(k0 + kk + kb + 1) * N];    // B[k+kb+1][n]
            acc = __builtin_amdgcn_wmma_f32_16x16x4_f32(
                /*neg_a=*/false, a, /*neg_b=*/false, b,
                /*c_mod=*/(short)0, acc, /*reuse_a=*/false, /*reuse_b=*/false);
        }
    }

    const float bv = bias ? bias[col0 + m] : 0.0f;
    #pragma unroll
    for (int r = 0; r < 8; ++r) {
        int mm = r + half * 8;
        float v = acc[r] + bv;
        if (act) v = v > 0.0f ? v : v * slope;
        C[(size_t)(row0 + mm) * N + col0 + m] = v;
    }
}

// ---------------------------------------------------------------------------
// Per-edge attention logits.  One wave per (edge, head); C = 128 channels,
// 4 channels per lane, wave-shuffle reduction.
// ---------------------------------------------------------------------------
__global__ void edge_alpha_kernel(const float* __restrict__ h,
                                  const int* __restrict__ src,
                                  const int* __restrict__ dst,
                                  const float* __restrict__ attL,
                                  const float* __restrict__ attR,
                                  float* __restrict__ alpha,
                                  unsigned* __restrict__ amax,
                                  int E, int heads) {
    const int C = NHID;
    int gw = (blockIdx.x * blockDim.x + threadIdx.x) >> 5;
    if (gw >= E * heads) return;
    int e  = gw / heads;
    int hd = gw - e * heads;
    int lane = threadIdx.x & 31;
    int s = src[e], d = dst[e];
    size_t stride = (size_t)heads * C;

    const float4 xj = *(const float4*)(h + (size_t)s * stride + hd * C + lane * 4);
    const float4 xi = *(const float4*)(h + (size_t)d * stride + hd * C + lane * 4);
    const float4 al = *(const float4*)(attL + hd * C + lane * 4);
    const float4 ar = *(const float4*)(attR + hd * C + lane * 4);

    float sd = xi.x * xj.x + xi.y * xj.y + xi.z * xj.z + xi.w * xj.w;
    float sl = xj.x * al.x + xj.y * al.y + xj.z * al.z + xj.w * al.w;
    float sr = xi.x * ar.x + xi.y * ar.y + xi.z * ar.z + xi.w * ar.w;
    #pragma unroll
    for (int off = 16; off > 0; off >>= 1) {
        sd += __shfl_xor(sd, off, 32);
        sl += __shfl_xor(sl, off, 32);
        sr += __shfl_xor(sr, off, 32);
    }
    if (lane == 0) {
        float sig = 1.0f / (1.0f + expf(-sd));
        float a = (sl + sr) * sig;
        a = a > 0.0f ? a : 0.2f * a;           // leaky_relu(0.2)
        alpha[gw] = a;
        atomicMax(&amax[(size_t)d * heads + hd], fenc(a));
    }
}

// exp(alpha - max) + segment-sum of denominators
__global__ void edge_exp_kernel(const float* __restrict__ alpha,
                                const unsigned* __restrict__ amax,
                                float* __restrict__ denom,
                                const int* __restrict__ dst,
                                float* __restrict__ ex,
                                int E, int heads) {
    int idx = blockIdx.x * blockDim.x + threadIdx.x;
    if (idx >= E * heads) return;
    int e  = idx / heads;
    int hd = idx - e * heads;
    int d  = dst[e];
    float mx = fdec(amax[(size_t)d * heads + hd]);
    float v  = expf(alpha[idx] - mx);
    ex[idx]  = v;
    atomicAdd(&denom[(size_t)d * heads + hd], v);
}

// out[dst] += x_j * (ex / denom[dst]) : one wave per (edge, head)
__global__ void edge_agg_kernel(const float* __restrict__ h,
                                const int* __restrict__ src,
                                const int* __restrict__ dst,
                                const float* __restrict__ ex,
                                const float* __restrict__ denom,
                                float* __restrict__ out,
                                int E, int heads) {
    const int C = NHID;
    int gw = (blockIdx.x * blockDim.x + threadIdx.x) >> 5;
    if (gw >= E * heads) return;
    int e  = gw / heads;
    int hd = gw - e * heads;
    int lane = threadIdx.x & 31;
    int s = src[e], d = dst[e];
    float w = ex[gw] / (denom[(size_t)d * heads + hd] + 1e-16f);
    size_t stride = (size_t)heads * C;
    const float4 xj = *(const float4*)(h + (size_t)s * stride + hd * C + lane * 4);
    float* o = out + (size_t)d * stride + hd * C + lane * 4;
    atomicAdd(o + 0, xj.x * w);
    atomicAdd(o + 1, xj.y * w);
    atomicAdd(o + 2, xj.z * w);
    atomicAdd(o + 3, xj.w * w);
}

// data = leaky_relu(data + bias[col], slope), in place
__global__ void bias_lrelu_kernel(float* __restrict__ data,
                                  const float* __restrict__ bias,
                                  int n, int cols, float slope) {
    int i = blockIdx.x * blockDim.x + threadIdx.x;
    if (i >= n) return;
    float v = data[i] + bias[i % cols];
    data[i] = v > 0.0f ? v : v * slope;
}

// segment_max over batch id
__global__ void pool_max_kernel(const float* __restrict__ h,
                                const int* __restrict__ batch,
                                unsigned* __restrict__ enc, int N, int C) {
    int i = blockIdx.x * blockDim.x + threadIdx.x;
    if (i >= N * C) return;
    int n = i / C, c = i - n * C;
    atomicMax(&enc[(size_t)batch[n] * C + c], fenc(h[i]));
}

__global__ void pool_decode_kernel(const unsigned* enc, float* out, int n) {
    int i = blockIdx.x * blockDim.x + threadIdx.x;
    if (i < n) out[i] = fdec(enc[i]);
}

// ---------------------------------------------------------------------------
// host launcher
// ---------------------------------------------------------------------------
extern "C" void kernel_launch(void* const* d_in, const int* in_sizes, int n_in,
                              void* d_out, int out_size, void* d_ws, size_t ws_size,
                              hipStream_t stream) {
    (void)in_sizes; (void)n_in; (void)out_size; (void)ws_size;

    const float* x      = (const float*)d_in[0];
    const int*   ei     = (const int*)  d_in[1];
    const int*   batch  = (const int*)  d_in[2];
    const float* fc1_w  = (const float*)d_in[3];
    const float* fc1_b  = (const float*)d_in[4];
    const float* w1     = (const float*)d_in[5];
    const float* attl1  = (const float*)d_in[6];
    const float* attr1  = (const float*)d_in[7];
    const float* b1     = (const float*)d_in[8];
    const float* w2     = (const float*)d_in[9];
    const float* attl2  = (const float*)d_in[10];
    const float* attr2  = (const float*)d_in[11];
    const float* b2     = (const float*)d_in[12];
    const float* fc2_w  = (const float*)d_in[13];
    const float* fc2_b  = (const float*)d_in[14];
    float* out = (float*)d_out;

    const int* src = ei;            // edge_index[0]
    const int* dst = ei + N_EDGES;  // edge_index[1]

    // carve workspace (256B aligned slices)
    char* p = (char*)d_ws;
    auto carve = [&](size_t bytes) -> void* {
        void* r = (void*)p;
        p += (bytes + 255) & ~(size_t)255;
        return r;
    };
    float*    H0   = (float*)   carve((size_t)N_NODES * NHID * 4);           // fc1 out
    float*    H1   = (float*)   carve((size_t)N_NODES * HEADS * NHID * 4);   // h0 @ w1
    float*    AG1  = (float*)   carve((size_t)N_NODES * HEADS * NHID * 4);   // conv1 agg
    float*    H2   = (float*)   carve((size_t)N_NODES * NHID * 4);           // h1 @ w2
    float*    AG2  = (float*)   carve((size_t)N_NODES * NHID * 4);           // conv2 agg
    float*    AL1  = (float*)   carve((size_t)N_EDGES * HEADS * 4);
    float*    EX1  = (float*)   carve((size_t)N_EDGES * HEADS * 4);
    unsigned* AM1  = (unsigned*)carve((size_t)N_NODES * HEADS * 4);
    float*    DN1  = (float*)   carve((size_t)N_NODES * HEADS * 4);
    float*    AL2  = (float*)   carve((size_t)N_EDGES * 4);
    float*    EX2  = (float*)   carve((size_t)N_EDGES * 4);
    unsigned* AM2  = (unsigned*)carve((size_t)N_NODES * 4);
    float*    DN2  = (float*)   carve((size_t)N_NODES * 4);
    unsigned* PENC = (unsigned*)carve((size_t)NGRAPHS * NHID * 4);
    float*    POOL = (float*)   carve((size_t)NGRAPHS * NHID * 4);

    const int TB = 256;
    auto blocks = [](int n, int tb) { return (n + tb - 1) / tb; };

    // 1) h0 = leaky_relu(x @ fc1_w + fc1_b, 0.01)   [20000 x 128]
    gemm16_wmma<<<dim3(N_NODES / 16, (NHID / 16) / 8), TB, 0, stream>>>(
        x, fc1_w, fc1_b, H0, N_NODES, NHID, N_IN, 0.01f, 1);

    // 2) h1 = h0 @ w1                                [20000 x 512]
    gemm16_wmma<<<dim3(N_NODES / 16, (HEADS * NHID / 16) / 8), TB, 0, stream>>>(
        H0, w1, nullptr, H1, N_NODES, HEADS * NHID, NHID, 0.0f, 0);

    // 3) conv1 attention (4 heads)
    fill_u32_kernel<<<blocks(N_NODES * HEADS, TB), TB, 0, stream>>>(AM1, ENC_NEG_INF, N_NODES * HEADS);
    fill_f32_kernel<<<blocks(N_NODES * HEADS, TB), TB, 0, stream>>>(DN1, 0.0f, N_NODES * HEADS);
    fill_f32_kernel<<<blocks(N_NODES * HEADS * NHID, TB), TB, 0, stream>>>(AG1, 0.0f, N_NODES * HEADS * NHID);
    edge_alpha_kernel<<<(N_EDGES * HEADS) / 8, TB, 0, stream>>>(
        H1, src, dst, attl1, attr1, AL1, AM1, N_EDGES, HEADS);
    edge_exp_kernel<<<blocks(N_EDGES * HEADS, TB), TB, 0, stream>>>(
        AL1, AM1, DN1, dst, EX1, N_EDGES, HEADS);
    edge_agg_kernel<<<(N_EDGES * HEADS) / 8, TB, 0, stream>>>(
        H1, src, dst, EX1, DN1, AG1, N_EDGES, HEADS);
    bias_lrelu_kernel<<<blocks(N_NODES * HEADS * NHID, TB), TB, 0, stream>>>(
        AG1, b1, N_NODES * HEADS * NHID, HEADS * NHID, 0.01f);

    // 4) h2 = h1f @ w2                               [20000 x 128]
    gemm16_wmma<<<dim3(N_NODES / 16, (NHID / 16) / 8), TB, 0, stream>>>(
        AG1, w2, nullptr, H2, N_NODES, NHID, HEADS * NHID, 0.0f, 0);

    // 5) conv2 attention (1 head)
    fill_u32_kernel<<<blocks(N_NODES, TB), TB, 0, stream>>>(AM2, ENC_NEG_INF, N_NODES);
    fill_f32_kernel<<<blocks(N_NODES, TB), TB, 0, stream>>>(DN2, 0.0f, N_NODES);
    fill_f32_kernel<<<blocks(N_NODES * NHID, TB), TB, 0, stream>>>(AG2, 0.0f, N_NODES * NHID);
    edge_alpha_kernel<<<N_EDGES / 8, TB, 0, stream>>>(
        H2, src, dst, attl2, attr2, AL2, AM2, N_EDGES, 1);
    edge_exp_kernel<<<blocks(N_EDGES, TB), TB, 0, stream>>>(
        AL2, AM2, DN2, dst, EX2, N_EDGES, 1);
    edge_agg_kernel<<<N_EDGES / 8, TB, 0, stream>>>(
        H2, src, dst, EX2, DN2, AG2, N_EDGES, 1);
    bias_lrelu_kernel<<<blocks(N_NODES * NHID, TB), TB, 0, stream>>>(
        AG2, b2, N_NODES * NHID, NHID, 0.01f);

    // 6) graph pooling: segment_max over batch       [128 x 128]
    fill_u32_kernel<<<blocks(NGRAPHS * NHID, TB), TB, 0, stream>>>(PENC, ENC_NEG_INF, NGRAPHS * NHID);
    pool_max_kernel<<<blocks(N_NODES * NHID, TB), TB, 0, stream>>>(AG2, batch, PENC, N_NODES, NHID);
    pool_decode_kernel<<<blocks(NGRAPHS * NHID, TB), TB, 0, stream>>>(PENC, POOL, NGRAPHS * NHID);

    // 7) out = pooled @ fc2_w + fc2_b                [128 x 768]
    gemm16_wmma<<<dim3(NGRAPHS / 16, (NOUT / 16) / 8), TB, 0, stream>>>(
        POOL, fc2_w, fc2_b, out, NGRAPHS, NOUT, NHID, 0.0f, 0);
}